// HierarchyModel_64647847739587
// MI455X (gfx1250) — compile-verified
//
#include <hip/hip_runtime.h>
#include <math.h>

// ---------------------------------------------------------------------------
// Problem constants (match reference)
// ---------------------------------------------------------------------------
#define N_SEL 2048
#define DIM   16
#define HEMB  128
#define EPS_F 1e-10f
#define MTILE 64          // 32x32 macro-tiles per side (N_SEL/32)
#define NTRI  (MTILE * (MTILE + 1) / 2)   // 2080 upper-triangular macro-tiles

typedef __attribute__((ext_vector_type(2))) float v2f;
typedef __attribute__((ext_vector_type(4))) float v4f;
typedef __attribute__((ext_vector_type(8))) float v8f;

// ---------------------------------------------------------------------------
// Workspace layout (float offsets)
// ---------------------------------------------------------------------------
#define WS_CL   0                        // gathered childrenLower  [N,D]
#define WS_CH   (WS_CL + N_SEL * DIM)    // gathered childrenHigher [N,D]
#define WS_SQ   (WS_CH + N_SEL * DIM)    // ||omega_i||^2            [N]
#define WS_ROWO (WS_SQ + N_SEL)          // row sums of omegaDist    [N]
#define WS_ROWR (WS_ROWO + N_SEL)        // row sums of realDist     [N]
#define WS_SCAL (WS_ROWR + N_SEL)        // scalar accumulators      [32]
// scalars: 0 sumSq(omegaDist), 1 sumSq(realDist), 2 lossExceed,
//          3 lossShapeLike,    4 lossOverlap,     8+d diffColSum[d]
#define WS_ZERO_COUNT (2 * N_SEL + 32)

// ---------------------------------------------------------------------------
// wave32 reductions
// ---------------------------------------------------------------------------
__device__ inline float waveReduce32(float v) {
#pragma unroll
  for (int off = 16; off > 0; off >>= 1) v += __shfl_xor(v, off, 32);
  return v;
}
// reduce across the 16 lanes sharing (lane>>4)
__device__ inline float halfReduce16(float v) {
#pragma unroll
  for (int off = 8; off > 0; off >>= 1) v += __shfl_xor(v, off, 32);
  return v;
}

// ---------------------------------------------------------------------------
// Kernel 0: zero the accumulator region of the workspace
// ---------------------------------------------------------------------------
__global__ void k_init(float* __restrict__ ws) {
  int i = blockIdx.x * blockDim.x + threadIdx.x;
  if (i < WS_ZERO_COUNT) ws[WS_ROWO + i] = 0.f;
}

// ---------------------------------------------------------------------------
// Kernel 1: gather boxes + elementwise losses + ||omega||^2
// ---------------------------------------------------------------------------
__global__ void k_gather(const int* __restrict__ idIdx,
                         const float* __restrict__ omega,
                         const float* __restrict__ chL,
                         const float* __restrict__ chH,
                         const float* __restrict__ pL,
                         const float* __restrict__ pH,
                         const float* __restrict__ pR,
                         const float* __restrict__ lr,
                         float* __restrict__ ws) {
  const int i = blockIdx.x * blockDim.x + threadIdx.x;   // grid sized exactly
  const int id = idIdx[i];
  const float ratio = lr[id];
  const float PI_2 = 1.57079632679489662f;

  float exceed = 0.f, shape = 0.f;
#pragma unroll
  for (int d = 0; d < DIM; ++d) {
    float cl = chL[id * DIM + d];
    float ch = chH[id * DIM + d];
    ws[WS_CL + i * DIM + d] = cl;
    ws[WS_CH + i * DIM + d] = ch;
    float l = pL[d], h = pH[d];
    exceed += fmaxf(l - cl, 0.f) + fmaxf(ch - h, 0.f) +
              fmaxf(l - ch, 0.f) + fmaxf(cl - h, 0.f);
    float diff = ch - cl;
    atomicAdd(&ws[WS_SCAL + 8 + d], diff);               // per-dim column sum
    float numer = fmaxf(diff / pR[d], EPS_F);
    float sd = fminf(fmaxf(numer / ratio, 0.01f), 1.99f);
    shape += fabsf(tanf((sd - 1.f) * PI_2));
  }

  // squared norm of omega row i (128 floats, 32x float4)
  float sq = 0.f;
  const v4f* om = (const v4f*)(omega + (size_t)i * HEMB);
#pragma unroll 4
  for (int h = 0; h < HEMB / 4; ++h) {
    v4f v = om[h];
    sq += v.x * v.x + v.y * v.y + v.z * v.z + v.w * v.w;
  }
  ws[WS_SQ + i] = sq;

  float eR = waveReduce32(exceed);
  float sR = waveReduce32(shape);
  if ((threadIdx.x & 31) == 0) {
    atomicAdd(&ws[WS_SCAL + 2], eR);
    atomicAdd(&ws[WS_SCAL + 3], sR);
  }
}

// ---------------------------------------------------------------------------
// Kernel 2: omegaDist via WMMA f32 16x16x4, symmetric 32x32 macro-tiles.
//   One wave = one 32x32 macro-tile (2x2 sub-tiles, 4 accumulators) on the
//   upper triangle. 4 frag loads -> 4 WMMAs per K-step (1:1 wmma:vmem).
//   Off-diagonal tiles: double Frobenius contribution and scatter od into
//   both rowSum[i] and rowSum[j] (matrix is symmetric, colSum == rowSum).
// ---------------------------------------------------------------------------
__global__ void k_omega(const float* __restrict__ omega,
                        float* __restrict__ ws) {
  const int lane = threadIdx.x & 31;
  const int wid  = blockIdx.x * (blockDim.x >> 5) + (threadIdx.x >> 5);

  // upper-triangular decode over MTILE x MTILE macro-grid (wave-uniform)
  int ti = (int)((2 * MTILE + 1 -
                  sqrtf((float)((2 * MTILE + 1) * (2 * MTILE + 1) - 8 * wid))) *
                 0.5f);
  if (ti < 0) ti = 0;
  while (ti > 0 && ti * (2 * MTILE - ti + 1) / 2 > wid) --ti;
  while ((ti + 1) * (2 * MTILE - ti) / 2 <= wid) ++ti;
  const int tj = ti + (wid - ti * (2 * MTILE - ti + 1) / 2);
  const int Ib = ti * 32, Jb = tj * 32;
  const bool offdiag = (ti != tj);
  const float wgt = offdiag ? 2.f : 1.f;

  // A frag: 16x4 f32. lanes 0-15: M=lane, K={k0,k0+1}; lanes 16-31: K={k0+2,k0+3}
  const int r16   = lane & 15;
  const int khalf = (lane >> 4) << 1;
  const float* A0 = omega + (size_t)(Ib + r16) * HEMB + khalf;
  const float* A1 = omega + (size_t)(Ib + 16 + r16) * HEMB + khalf;
  const float* B0 = omega + (size_t)(Jb + r16) * HEMB + khalf;
  const float* B1 = omega + (size_t)(Jb + 16 + r16) * HEMB + khalf;

  __builtin_prefetch(A0, 0, 1);
  __builtin_prefetch(B0, 0, 1);

  v8f c00 = {}, c01 = {}, c10 = {}, c11 = {};
#pragma unroll 4
  for (int k0 = 0; k0 < HEMB; k0 += 4) {
    v2f a0 = *(const v2f*)(A0 + k0);
    v2f a1 = *(const v2f*)(A1 + k0);
    v2f b0 = *(const v2f*)(B0 + k0);
    v2f b1 = *(const v2f*)(B1 + k0);
    c00 = __builtin_amdgcn_wmma_f32_16x16x4_f32(false, a0, false, b0,
                                                (short)0, c00, false, false);
    c01 = __builtin_amdgcn_wmma_f32_16x16x4_f32(false, a0, false, b1,
                                                (short)0, c01, false, false);
    c10 = __builtin_amdgcn_wmma_f32_16x16x4_f32(false, a1, false, b0,
                                                (short)0, c10, false, false);
    c11 = __builtin_amdgcn_wmma_f32_16x16x4_f32(false, a1, false, b1,
                                                (short)0, c11, false, false);
  }
  v8f acc[2][2] = {{c00, c01}, {c10, c11}};

  // D layout per sub-tile: lane = column (lane&15); VGPR r = row r + 8*(lane>>4)
  float ssq = 0.f;
#pragma unroll
  for (int a = 0; a < 2; ++a) {
    const int ibase = Ib + a * 16 + ((lane >> 4) << 3);
#pragma unroll
    for (int b = 0; b < 2; ++b) {
      const int jl = Jb + b * 16 + r16;
      const float sqj = ws[WS_SQ + jl];
      float cacc = 0.f;
#pragma unroll
      for (int r = 0; r < 8; ++r) {
        const int i = ibase + r;
        const float sqi = ws[WS_SQ + i];
        float od = fmaxf(fmaf(-2.f, acc[a][b][r], sqi + sqj), EPS_F);
        ssq += wgt * od * od;
        cacc += od;
        float rsum = halfReduce16(od);           // sum over the 16 columns
        if (r16 == 0) atomicAdd(&ws[WS_ROWO + i], rsum);
      }
      if (offdiag) {                             // mirror: colSum == rowSum[j]
        cacc += __shfl_xor(cacc, 16, 32);
        if (lane < 16) atomicAdd(&ws[WS_ROWO + Jb + b * 16 + lane], cacc);
      }
    }
  }
  float t = waveReduce32(ssq);
  if (lane == 0) atomicAdd(&ws[WS_SCAL + 0], t);
}

// ---------------------------------------------------------------------------
// Kernel 3: box pairwise pass — lossOverlap + realDist row-sums / sumSq.
//   Block = 256 threads; tile = 16 rows x 256 cols. Rows staged in LDS.
// ---------------------------------------------------------------------------
__global__ void k_box(float* __restrict__ ws) {
  __shared__ float s_cl[16][DIM];
  __shared__ float s_ch[16][DIM];
  const int Ib = blockIdx.x * 16;
  const int Jb = blockIdx.y * 256;
  const int t  = threadIdx.x;

  {
    int rr = t >> 4, dd = t & 15;
    s_cl[rr][dd] = ws[WS_CL + (Ib + rr) * DIM + dd];
    s_ch[rr][dd] = ws[WS_CH + (Ib + rr) * DIM + dd];
  }
  const int j = Jb + t;
  float mycl[DIM], mych[DIM];
#pragma unroll
  for (int d = 0; d < DIM; ++d) {
    mycl[d] = ws[WS_CL + j * DIM + d];
    mych[d] = ws[WS_CH + j * DIM + d];
  }
  __syncthreads();

  float ssq = 0.f, ov = 0.f;
  for (int i = 0; i < 16; ++i) {
    const int gi = Ib + i;
    float l1 = 0.f, o = 0.f;
#pragma unroll
    for (int d = 0; d < DIM; ++d) {
      float a = s_cl[i][d], b = s_ch[i][d];
      o  += fmaxf(fminf(b, mych[d]) - fmaxf(a, mycl[d]), 0.f);
      l1 += fabsf(a - mycl[d]);
    }
    if (gi != j) ov += o;      // offdiag mask (l1 diag == 0 naturally)
    ssq += l1 * l1;
    float rs = waveReduce32(l1);
    if ((t & 31) == 0) atomicAdd(&ws[WS_ROWR + gi], rs);
  }
  float s1 = waveReduce32(ssq);
  float s2 = waveReduce32(ov);
  if ((t & 31) == 0) {
    atomicAdd(&ws[WS_SCAL + 1], s1);
    atomicAdd(&ws[WS_SCAL + 4], s2);
  }
}

// ---------------------------------------------------------------------------
// Kernel 4: finalize — assemble the scalar loss.
//   ||A/||A|| - d||_F^2 = SA2/nO^2 - (2/nO)*sum_j d_j*rowO_j + N*sum_j d_j^2
// ---------------------------------------------------------------------------
__global__ void k_final(const float* __restrict__ pR,
                        const float* __restrict__ ws,
                        float* __restrict__ out) {
  __shared__ float sh2[256], sh3[256];
  const int t = threadIdx.x;
  const float sumSqO = ws[WS_SCAL + 0];
  const float sumSqR = ws[WS_SCAL + 1];
  const float normO = fmaxf(sqrtf(sumSqO), EPS_F);
  const float normR = fmaxf(sqrtf(sumSqR), EPS_F);

  float a2 = 0.f, a3 = 0.f;
  for (int jj = t; jj < N_SEL; jj += 256) {
    float dj = ws[WS_ROWR + jj] / normR;
    a2 += dj * ws[WS_ROWO + jj];
    a3 += dj * dj;
  }
  sh2[t] = a2; sh3[t] = a3;
  __syncthreads();
  for (int s = 128; s > 0; s >>= 1) {
    if (t < s) { sh2[t] += sh2[t + s]; sh3[t] += sh3[t + s]; }
    __syncthreads();
  }
  if (t == 0) {
    float t1 = sumSqO / (normO * normO);
    float lossDist =
        sqrtf(fmaxf(t1 - 2.f * sh2[0] / normO + (float)N_SEL * sh3[0], 0.f));
    float lossPos = 0.f;
#pragma unroll
    for (int d = 0; d < DIM; ++d)
      lossPos += fmaxf(pR[d] - ws[WS_SCAL + 8 + d], 0.f);
    out[0] = lossDist + ws[WS_SCAL + 3] + ws[WS_SCAL + 2] +
             ws[WS_SCAL + 4] + lossPos;
  }
}

// ---------------------------------------------------------------------------
// Host-side launch
// ---------------------------------------------------------------------------
extern "C" void kernel_launch(void* const* d_in, const int* in_sizes, int n_in,
                              void* d_out, int out_size, void* d_ws,
                              size_t ws_size, hipStream_t stream) {
  const int*   idIdx = (const int*)d_in[0];
  const float* omega = (const float*)d_in[1];
  // d_in[2] = epoch (unused)
  const float* chL = (const float*)d_in[3];
  const float* chH = (const float*)d_in[4];
  const float* pL  = (const float*)d_in[5];
  const float* pH  = (const float*)d_in[6];
  const float* pR  = (const float*)d_in[7];
  const float* lr  = (const float*)d_in[8];
  float* ws  = (float*)d_ws;
  float* out = (float*)d_out;

  k_init<<<(WS_ZERO_COUNT + 255) / 256, 256, 0, stream>>>(ws);
  k_gather<<<N_SEL / 256, 256, 0, stream>>>(idIdx, omega, chL, chH, pL, pH, pR,
                                            lr, ws);
  // 2080 upper-triangular 32x32 macro-tiles, 8 waves per 256-thread block
  k_omega<<<NTRI / 8, 256, 0, stream>>>(omega, ws);
  dim3 gbox(N_SEL / 16, N_SEL / 256);
  k_box<<<gbox, 256, 0, stream>>>(ws);
  k_final<<<1, 256, 0, stream>>>(pR, ws, out);
}